// TextualGraph_18622978196311
// MI455X (gfx1250) — compile-verified
//
#include <hip/hip_runtime.h>
#include <hip/hip_bf16.h>

typedef float v2f __attribute__((ext_vector_type(2)));
typedef float v8f __attribute__((ext_vector_type(8)));

#define NI 64
#define NR 36          // regions
#define D  1024        // embed dim
#define NC 64
#define LMAX 40        // max caption length
#define NB 32          // num blocks
#define BS 32          // block size = D/NB
#define OUT_DIM 6
#define HID 8
#define LAM 20.0f
#define EPSF 1e-8f

#define KC 64          // K-chunk (GEMM1) / N-chunk (GEMM2)
#define MT 48          // padded regions (36 -> 48)
#define NT 48          // padded words   (40 -> 48)
#define STRA 68        // LDS stride (floats): 272B rows -> 16B aligned, bank-spread
#define STRT 49        // LDS stride for attn^T

// Issue an async global->LDS 128-bit copy (tracked by ASYNCcnt).
__device__ __forceinline__ void async_copy_b128(void* lds, const void* gptr) {
    unsigned ldsOff = (unsigned)(uintptr_t)lds;   // LDS aperture: addr[31:0] = LDS byte offset
    asm volatile("global_load_async_to_lds_b128 %0, %1, off"
                 :: "v"(ldsOff), "v"(gptr) : "memory");
}
__device__ __forceinline__ void wait_async0() {
    asm volatile("s_wait_asynccnt 0x0" ::: "memory");
}

__global__ __launch_bounds__(288) void scan_sim_kernel(
    const float* __restrict__ images, const float* __restrict__ captions,
    const int* __restrict__ cap_lens, const float* __restrict__ rare,
    const float* __restrict__ v1, const float* __restrict__ g1, const float* __restrict__ b1,
    const float* __restrict__ v2, const float* __restrict__ g2, const float* __restrict__ b2,
    float* __restrict__ out)
{
    __shared__ __align__(16) float sA[MT * STRA]; // GEMM1 A chunk (image rows); wctx tile in GEMM2
    __shared__ __align__(16) float sB[NT * STRA]; // GEMM1 B chunk (caption rows); image col-chunk
    __shared__ __align__(16) float sT[NT * STRT]; // attn^T [word l][region r]
    __shared__ float cosb[LMAX * NB];             // blockwise cosine per (word, block)
    __shared__ float w1s[HID * NB];
    __shared__ float w2s[OUT_DIM * HID];
    __shared__ float colInv[NR];
    __shared__ float red[LMAX];

    const int i    = blockIdx.x;
    const int c    = blockIdx.y;
    const int tid  = threadIdx.x;
    const int lane = tid & 31;
    const int wave = tid >> 5;        // 0..8
    const int half = lane >> 4;       // 0/1 lane halves
    const int l15  = lane & 15;

    const float* img = images  + (size_t)i * NR * D;
    const float* cap = captions + (size_t)c * LMAX * D;
    const int    len = cap_lens[c];

    // ---- one-time zero fill of pad rows (regions 36..47 / words 40..47) ----
    for (int idx = tid; idx < (MT - NR) * KC; idx += 288)
        sA[(NR + idx / KC) * STRA + (idx % KC)] = 0.f;
    for (int idx = tid; idx < (NT - LMAX) * KC; idx += 288)
        sB[(LMAX + idx / KC) * STRA + (idx % KC)] = 0.f;

    // ---- weight_norm (dim=0, torch default): w[j,:] = g[j]*v[j,:]/||v[j,:]|| ----
    if (tid < HID) {
        float s = 0.f;
        for (int k = 0; k < NB; ++k) { float x = v1[tid * NB + k]; s += x * x; }
        float inv = g1[tid] * rsqrtf(s);
        for (int k = 0; k < NB; ++k) w1s[tid * NB + k] = v1[tid * NB + k] * inv;
    } else if (tid < HID + OUT_DIM) {
        int j = tid - HID;
        float s = 0.f;
        for (int k = 0; k < HID; ++k) { float x = v2[j * HID + k]; s += x * x; }
        float inv = g2[j] * rsqrtf(s);
        for (int k = 0; k < HID; ++k) w2s[j * HID + k] = v2[j * HID + k] * inv;
    }

    // =========== GEMM1: attn[m=region, n=word] = img · cap^T over K=D ===========
    const int mt = wave / 3, nt = wave % 3;   // 3x3 tile grid over 48x48
    v8f acc = {};
    for (int k0 = 0; k0 < D; k0 += KC) {
        // async-stage image rows (36 x 64) and caption rows (40 x 64), 16B vectors
        for (int v = tid; v < NR * (KC / 4); v += 288) {
            int m = v >> 4, kq = (v & 15) << 2;
            async_copy_b128(&sA[m * STRA + kq], &img[m * D + k0 + kq]);
        }
        for (int v = tid; v < LMAX * (KC / 4); v += 288) {
            int n = v >> 4, kq = (v & 15) << 2;
            async_copy_b128(&sB[n * STRA + kq], &cap[n * D + k0 + kq]);
        }
        if (k0 + KC < D && tid < NR) __builtin_prefetch(&img[tid * D + k0 + KC], 0, 0);
        wait_async0();
        __syncthreads();
        const float* ap = &sA[(mt * 16 + l15) * STRA + 2 * half];
        const float* bp = &sB[(nt * 16 + l15) * STRA + 2 * half];
        #pragma unroll
        for (int kk = 0; kk < KC; kk += 4) {
            // A 16x4 f32 layout: v0 = K0 (lanes0-15)/K2 (lanes16-31), v1 = K1/K3
            v2f a; a.x = ap[kk]; a.y = ap[kk + 1];
            v2f b; b.x = bp[kk]; b.y = bp[kk + 1];
            acc = __builtin_amdgcn_wmma_f32_16x16x4_f32(false, a, false, b,
                                                        (short)0, acc, false, false);
        }
        __syncthreads();   // DS reads done before next chunk's async writes
    }

    // ---- epilogue: LeakyReLU(0.1) * word-mask, transpose into sT[l][r] ----
    #pragma unroll
    for (int v = 0; v < 8; ++v) {
        int m = mt * 16 + v + 8 * half;  // region (C layout: VGPR v -> M=v / v+8)
        int n = nt * 16 + l15;           // word
        float val = acc[v];
        val = (val > 0.f) ? val : 0.1f * val;
        if (m >= NR || n >= len) val = 0.f;
        sT[n * STRT + m] = val;
    }
    __syncthreads();

    // ---- per-region L2 norm over words ----
    if (tid < NR) {
        float s = 0.f;
        for (int l = 0; l < LMAX; ++l) { float x = sT[l * STRT + tid]; s += x * x; }
        colInv[tid] = 1.f / (sqrtf(s) + EPSF);
    }
    __syncthreads();

    // ---- softmax over regions (×lambda), per word row; rows >= LMAX stay zero ----
    if (tid < LMAX) {
        float mx = -1e30f;
        for (int r = 0; r < NR; ++r) {
            float x = sT[tid * STRT + r] * colInv[r] * LAM;
            mx = fmaxf(mx, x);
        }
        float s = 0.f;
        for (int r = 0; r < NR; ++r) {
            float e = __expf(sT[tid * STRT + r] * colInv[r] * LAM - mx);
            sT[tid * STRT + r] = e; s += e;
        }
        float inv = 1.f / s;
        for (int r = 0; r < NR; ++r) sT[tid * STRT + r] *= inv;
    }
    __syncthreads();

    // ==== GEMM2: wctx[l, d-chunk] = attn^T (40x36) · img (36xD), fused blockwise cosine ====
    for (int c0 = 0; c0 < D; c0 += KC) {
        for (int v = tid; v < NR * (KC / 4); v += 288) {
            int r = v >> 4, nq = (v & 15) << 2;
            async_copy_b128(&sB[r * STRA + nq], &img[r * D + c0 + nq]);
        }
        wait_async0();
        __syncthreads();
        // 12 tile jobs (3 word-tiles x 4 col-tiles) over 9 waves
        for (int job = wave; job < 12; job += 9) {
            int m2 = job >> 2;   // word tile 0..2
            int n2 = job & 3;    // col tile  0..3
            v8f w = {};
            const float* ap = &sT[(m2 * 16 + l15) * STRT + 2 * half];
            const float* bp = &sB[(2 * half) * STRA + n2 * 16 + l15];
            #pragma unroll
            for (int kk = 0; kk < NR; kk += 4) {   // K = 36 regions, exactly 9 steps
                v2f a; a.x = ap[kk];        a.y = ap[kk + 1];
                v2f b; b.x = bp[kk * STRA]; b.y = bp[(kk + 1) * STRA];
                w = __builtin_amdgcn_wmma_f32_16x16x4_f32(false, a, false, b,
                                                          (short)0, w, false, false);
            }
            #pragma unroll
            for (int v = 0; v < 8; ++v) {
                int mm = m2 * 16 + v + 8 * half;
                int nn = n2 * 16 + l15;
                sA[mm * STRA + nn] = w[v];   // wctx tile into sA (freed after GEMM1)
            }
        }
        __syncthreads();
        // blockwise cosine for the two 32-wide blocks in this 64-col chunk
        if (tid < LMAX * 2) {
            int l   = tid % LMAX;
            int nbl = tid / LMAX;             // 0..1
            int nbg = (c0 / BS) + nbl;        // global block index
            float w12 = 0.f, cn2 = 0.f, qn2 = 0.f;
            const float* wrow = &sA[l * STRA + nbl * BS];
            const float* qrow = &cap[l * D + c0 + nbl * BS];
            for (int b = 0; b < BS; ++b) {
                float wv = wrow[b], qv = qrow[b];
                w12 += wv * qv; cn2 += wv * wv; qn2 += qv * qv;
            }
            cosb[l * NB + nbg] = w12 / fmaxf(sqrtf(qn2) * sqrtf(cn2), EPSF);
        }
        // next iteration's post-load barrier protects sA/sB before they are rewritten
    }
    __syncthreads();

    // =========== tiny weight-normed MLP + masked mean ===========
    if (tid < LMAX) {
        float s = 0.f;
        if (tid < len) {
            const float* rr = rare + ((size_t)c * LMAX + tid) * NB;
            float h[HID];
            for (int j = 0; j < HID; ++j) {
                float a = b1[j];
                for (int nb = 0; nb < NB; ++nb)
                    a += (cosb[tid * NB + nb] + 0.4f * rr[nb]) * w1s[j * NB + nb];
                h[j] = tanhf(a);
            }
            for (int o = 0; o < OUT_DIM; ++o) {
                float a = b2[o];
                for (int j = 0; j < HID; ++j) a += h[j] * w2s[o * HID + j];
                s += a;
            }
        }
        red[tid] = s;
    }
    __syncthreads();
    if (tid == 0) {
        float t = 0.f;
        for (int l = 0; l < LMAX; ++l) t += red[l];
        out[i * NC + c] = t / ((float)len * (float)OUT_DIM);
    }
}

extern "C" void kernel_launch(void* const* d_in, const int* in_sizes, int n_in,
                              void* d_out, int out_size, void* d_ws, size_t ws_size,
                              hipStream_t stream) {
    (void)in_sizes; (void)n_in; (void)out_size; (void)d_ws; (void)ws_size;
    const float* images   = (const float*)d_in[0];
    const float* captions = (const float*)d_in[1];
    const int*   cap_lens = (const int*)d_in[2];
    const float* rare     = (const float*)d_in[3];
    const float* v1       = (const float*)d_in[4];
    const float* g1       = (const float*)d_in[5];
    const float* b1       = (const float*)d_in[6];
    const float* v2       = (const float*)d_in[7];
    const float* g2       = (const float*)d_in[8];
    const float* b2       = (const float*)d_in[9];
    float* out = (float*)d_out;

    dim3 grid(NI, NC);
    scan_sim_kernel<<<grid, 288, 0, stream>>>(images, captions, cap_lens, rare,
                                              v1, g1, b1, v2, g2, b2, out);
}